// GraphSAGE_13615046328531
// MI455X (gfx1250) — compile-verified
//
#include <hip/hip_runtime.h>
#include <hip/hip_bf16.h>

// GraphSAGE (mean) x2 layers, 50000 nodes, 600000 edges, 128->128->128.
//
// Roofline: h = 25.6MB -> fully L2-resident (192MB L2). Aggregation moves
// ~1.2GB through L2 (gather + 76.8M f32 atomic-adds per layer) -> L2/atomic
// bound, HBM mostly idle. GEMMs = 6.55 GFLOP fp32 total -> trivial on the
// matrix pipe; V_WMMA_F32_16X16X4_F32 keeps full fp32 precision (reference
// is fp32), and at this size precision > peak-rate is the right trade.

#define NFEAT 128

typedef float v2f __attribute__((ext_vector_type(2)));
typedef float v8f __attribute__((ext_vector_type(8)));

__global__ void fill_zero4_kernel(float4* __restrict__ p, long n4) {
    long i = (long)blockIdx.x * blockDim.x + threadIdx.x;
    if (i < n4) p[i] = make_float4(0.f, 0.f, 0.f, 0.f);
}

__global__ void degree_kernel(const int* __restrict__ dst, float* __restrict__ deg, int nedges) {
    int e = blockIdx.x * blockDim.x + threadIdx.x;
    if (e < nedges) {
        __hip_atomic_fetch_add(&deg[dst[e]], 1.0f, __ATOMIC_RELAXED, __HIP_MEMORY_SCOPE_AGENT);
    }
}

__global__ void invdeg_kernel(float* __restrict__ deg, int n) {
    int i = blockIdx.x * blockDim.x + threadIdx.x;
    if (i < n) deg[i] = 1.0f / fmaxf(deg[i], 1.0f);
}

// One wave32 per edge: each lane carries float4 (lane*4 .. lane*4+3) of the
// 128-float feature row. Gather from feat[src], scatter into agg[dst] using
// native device-scope f32 atomics (guaranteed global_atomic_add_f32 -- the
// agg buffer is coarse-grained workspace memory, so the HW fp atomic is safe;
// SCOPE_DEV is required for cross-WGP RMW coherence).
__global__ void scatter_add_kernel(const float* __restrict__ feat,
                                   const int* __restrict__ src,
                                   const int* __restrict__ dst,
                                   float* __restrict__ agg, int nedges) {
    long gid  = (long)blockIdx.x * blockDim.x + threadIdx.x;
    int  e    = (int)(gid >> 5);
    int  lane = threadIdx.x & 31;
    if (e >= nedges) return;
    int s = src[e];
    int d = dst[e];
    const float4 v = ((const float4*)(feat + (long)s * NFEAT))[lane];
    float* o = agg + (long)d * NFEAT + lane * 4;
    asm volatile(
        "global_atomic_add_f32 %0, %1, off scope:SCOPE_DEV\n\t"
        "global_atomic_add_f32 %0, %2, off offset:4 scope:SCOPE_DEV\n\t"
        "global_atomic_add_f32 %0, %3, off offset:8 scope:SCOPE_DEV\n\t"
        "global_atomic_add_f32 %0, %4, off offset:12 scope:SCOPE_DEV"
        :: "v"(o), "v"(v.x), "v"(v.y), "v"(v.z), "v"(v.w)
        : "memory");
}

// out[m0:m0+16, :] = act( h @ Wself + diag(invdeg)*(agg @ Wneigh) + bias )
// 256 threads = 8 waves; wave w owns the 16x16 tile at columns [16w, 16w+16).
// Two accumulators (self/neigh) so the K-loop is pure loads + 2 WMMAs; the
// invdeg row-scaling commutes with the GEMM and is applied in the epilogue.
// Fragment layouts per CDNA5 ISA 7.12.2 (f32):
//   A (16x4, 2 VGPR): M = lane&15, K = v + 2*(lane>>4)
//   B (4x16, 2 VGPR): N = lane&15, K = v + 2*(lane>>4)
//   C (16x16, 8 VGPR): row = v + 8*(lane>>4), col = lane&15
template <bool RELU>
__global__ void sage_gemm_kernel(const float* __restrict__ h,
                                 const float* __restrict__ agg,
                                 const float* __restrict__ invdeg,
                                 const float* __restrict__ Wself,
                                 const float* __restrict__ Wneigh,
                                 const float* __restrict__ bias,
                                 float* __restrict__ out, int nrows) {
    const int lane = threadIdx.x & 31;
    const int wv   = threadIdx.x >> 5;       // 0..7
    const int n0   = wv * 16;                // output column tile
    const int m0   = blockIdx.x * 16;        // output row tile
    const int mn   = lane & 15;              // M for A, N for B/C
    const int hi   = lane >> 4;              // K/row-half selector

    int rowA = m0 + mn;
    if (rowA >= nrows) rowA = nrows - 1;     // clamp loads (tail tiles only)
    const long rbase = (long)rowA * NFEAT;

    v8f cS = {0.f, 0.f, 0.f, 0.f, 0.f, 0.f, 0.f, 0.f};
    v8f cN = {0.f, 0.f, 0.f, 0.f, 0.f, 0.f, 0.f, 0.f};

#pragma unroll 4
    for (int k = 0; k < NFEAT; k += 4) {
        v2f aS, aN, bS, bN;
#pragma unroll
        for (int v = 0; v < 2; ++v) {
            const int kk = k + v + 2 * hi;
            aS[v] = h[rbase + kk];
            aN[v] = agg[rbase + kk];
            bS[v] = Wself[(long)kk * NFEAT + n0 + mn];
            bN[v] = Wneigh[(long)kk * NFEAT + n0 + mn];
        }
        cS = __builtin_amdgcn_wmma_f32_16x16x4_f32(false, aS, false, bS, (short)0, cS, false, false);
        cN = __builtin_amdgcn_wmma_f32_16x16x4_f32(false, aN, false, bN, (short)0, cN, false, false);
    }

    const float bb = bias[n0 + mn];
    if (m0 + 16 <= nrows) {                  // uniform branch: full tile (hot path)
#pragma unroll
        for (int v = 0; v < 8; ++v) {
            const int row = m0 + v + 8 * hi;
            float val = fmaf(invdeg[row], cN[v], cS[v]) + bb;
            if (RELU) val = fmaxf(val, 0.0f);
            out[(long)row * NFEAT + n0 + mn] = val;
        }
    } else {                                 // tail tile (never taken for N=50000)
#pragma unroll
        for (int v = 0; v < 8; ++v) {
            const int row = m0 + v + 8 * hi;
            if (row < nrows) {
                float val = fmaf(invdeg[row], cN[v], cS[v]) + bb;
                if (RELU) val = fmaxf(val, 0.0f);
                out[(long)row * NFEAT + n0 + mn] = val;
            }
        }
    }
}

extern "C" void kernel_launch(void* const* d_in, const int* in_sizes, int n_in,
                              void* d_out, int out_size, void* d_ws, size_t ws_size,
                              hipStream_t stream) {
    const float* h   = (const float*)d_in[0];
    const int*   src = (const int*)  d_in[1];
    const int*   dst = (const int*)  d_in[2];
    const float* Ws0 = (const float*)d_in[3];
    const float* Wn0 = (const float*)d_in[4];
    const float* b0  = (const float*)d_in[5];
    const float* Ws1 = (const float*)d_in[6];
    const float* Wn1 = (const float*)d_in[7];
    const float* b1  = (const float*)d_in[8];
    float*       out = (float*)d_out;

    const int  nnodes = in_sizes[0] / NFEAT;
    const int  nedges = in_sizes[1];
    const long nf     = (long)nnodes * NFEAT;

    // Workspace layout (floats): [inv_deg | agg | h1]   (~51.5 MB)
    float* ws   = (float*)d_ws;
    float* invd = ws;
    long   off  = ((long)nnodes + 63) & ~63L;   // 256B-align next region
    float* agg  = ws + off;
    float* h1   = agg + nf;

    const int T = 256;
    const int gN  = (nnodes + T - 1) / T;
    const int gE  = (nedges + T - 1) / T;
    const int gNF4 = (int)((nf / 4 + T - 1) / T);
    const int gSC = (int)(((long)nedges * 32 + T - 1) / T);
    const int gMM = (nnodes + 15) / 16;

    // Degree (graph is static; compute once, reuse for both layers).
    // invd buffer doubles as the degree accumulator.
    fill_zero4_kernel<<<(int)((off / 4 + T - 1) / T), T, 0, stream>>>((float4*)invd, off / 4);
    degree_kernel<<<gE, T, 0, stream>>>(dst, invd, nedges);
    invdeg_kernel<<<gN, T, 0, stream>>>(invd, nnodes);

    // Layer 0: h1 = relu(h@Ws0 + mean@Wn0 + b0)
    fill_zero4_kernel<<<gNF4, T, 0, stream>>>((float4*)agg, nf / 4);
    scatter_add_kernel<<<gSC, T, 0, stream>>>(h, src, dst, agg, nedges);
    sage_gemm_kernel<true><<<gMM, T, 0, stream>>>(h, agg, invd, Ws0, Wn0, b0, h1, nnodes);

    // Layer 1: out = h1@Ws1 + mean@Wn1 + b1
    fill_zero4_kernel<<<gNF4, T, 0, stream>>>((float4*)agg, nf / 4);
    scatter_add_kernel<<<gSC, T, 0, stream>>>(h1, src, dst, agg, nedges);
    sage_gemm_kernel<false><<<gMM, T, 0, stream>>>(h1, agg, invd, Ws1, Wn1, b1, out, nnodes);
}